// VectorQuantizer_39797166964970
// MI455X (gfx1250) — compile-verified
//
#include <hip/hip_runtime.h>
#include <hip/hip_bf16.h>
#include <math.h>

typedef __attribute__((ext_vector_type(2))) float v2f;
typedef __attribute__((ext_vector_type(8))) float v8f;

// Problem constants
#define DDIM     256
#define KCODES   4096
#define NVEC     32768            // 32 * 32 * 32  (b*h*w)
#define ZELEMS   8388608          // 32*256*32*32
#define BSTRIDE  262144           // 256*1024 floats per batch image

// d_out layout (float element offsets, tuple return order)
#define OUT_ZQ      ((size_t)0)
#define OUT_LOSS    ((size_t)8388608)
#define OUT_PERP    ((size_t)8388609)
#define OUT_ONEHOT  ((size_t)8388610)                 // 32768*4096 floats
#define OUT_IDX     ((size_t)142606338)               // 8388610 + 134217728
#define OUT_DIST    ((size_t)142639106)               // OUT_IDX + 32768

// workspace layout (byte offsets)
#define WS_PARTIAL  0        // 2048 floats (per-block SSE partials)
#define WS_COUNTS   8192     // 4096 uint (codebook histogram)
#define WS_E2       24576    // 4096 floats (|e_k|^2)

// ---------------------------------------------------------------------------
// Zero the one-hot region (537 MB) + histogram + partials. b64 stores (the
// one-hot base is only 8B aligned).
__global__ void vq_zero(float* __restrict__ out,
                        unsigned int* __restrict__ counts,
                        float* __restrict__ partial) {
  const size_t n2 = (size_t)NVEC * KCODES / 2;   // 67108864 float2 slots
  v2f* p = (v2f*)(out + OUT_ONEHOT);
  const v2f z2 = {0.0f, 0.0f};
  size_t stride = (size_t)gridDim.x * blockDim.x;
  for (size_t i = (size_t)blockIdx.x * blockDim.x + threadIdx.x; i < n2; i += stride)
    p[i] = z2;
  size_t gid = (size_t)blockIdx.x * blockDim.x + threadIdx.x;
  if (gid < KCODES) counts[gid] = 0u;
  if (gid < 2048)   partial[gid] = 0.0f;
}

// ---------------------------------------------------------------------------
// |e_k|^2 : one wave per code row (coalesced), shuffle reduce (wave32).
__global__ void vq_e2(const float* __restrict__ E, float* __restrict__ e2) {
  const int lane = threadIdx.x & 31;
  const int wave = threadIdx.x >> 5;
  const int code = blockIdx.x * 8 + wave;
  const float* row = E + (size_t)code * DDIM;
  float s = 0.0f;
#pragma unroll
  for (int i = 0; i < 8; ++i) { float v = row[lane + i * 32]; s += v * v; }
  for (int off = 16; off > 0; off >>= 1) s += __shfl_xor(s, off, 32);
  if (lane == 0) e2[code] = s;
}

// ---------------------------------------------------------------------------
// Main: per block, a 16-row z tile vs all 4096 codes.
// 4 column tiles per A pass -> 4 independent WMMA accumulator chains (no
// serial RAW stall on the XDL pipe) and 4x less LDS read traffic.
__launch_bounds__(128)
__global__ void vq_main(const float* __restrict__ Z, const float* __restrict__ E,
                        const float* __restrict__ e2, float* __restrict__ out,
                        unsigned int* __restrict__ counts,
                        float* __restrict__ partial) {
  __shared__ float Atile[16 * 260];   // padded row stride -> conflict-free b64 reads
  __shared__ float z2s[16];
  __shared__ float argv[4 * 256];
  __shared__ int   argi[4 * 256];
  __shared__ int   idx_s[16];
  __shared__ float red[128];

  const int tid    = threadIdx.x;
  const int lane   = tid & 31;
  const int wave   = tid >> 5;
  const int rowbase = blockIdx.x << 4;       // 16 rows per block
  const int b       = rowbase >> 10;         // rows share one batch image
  const int hwbase  = rowbase & 1023;

  // Stage A tile (transpose for free: 16 rows are contiguous per channel d)
  const float* zb = Z + (size_t)b * BSTRIDE + hwbase;
  for (int i = tid; i < 4096; i += 128) {
    const int d = i >> 4, r = i & 15;
    Atile[r * 260 + d] = zb[(size_t)d * 1024 + r];
  }
  __syncthreads();
  if (tid < 16) {
    float s = 0.0f;
    for (int d = 0; d < DDIM; ++d) { float v = Atile[tid * 260 + d]; s += v * v; }
    z2s[tid] = s;
  }
  __syncthreads();

  // WMMA f32 16x16x4 fragment addressing:
  // A: v[0] = (M=lane%16, K = lane<16 ? 0 : 2), v[1] = K+1  -> one b64/lane
  // B: v[0] = (N=lane%16, K = lane<16 ? 0 : 2), v[1] = K+1  -> one b64/lane
  const int koff = (lane < 16) ? 0 : 2;
  const int nlo  = lane & 15;
  const float* Abase = Atile + nlo * 260 + koff;

  float minv[8];
  int   mini[8];
#pragma unroll
  for (int j = 0; j < 8; ++j) { minv[j] = 3.4e38f; mini[j] = 0x7fffffff; }

  float* dist_out = out + OUT_DIST;
  const int mo = (lane < 16) ? 0 : 8;          // C/D layout: vgpr j -> rows j / j+8

  // Wave w covers column tiles {w + 4k, k=0..63}; batch 4 tiles per pass:
  // tiles t0, t0+4, t0+8, t0+12  (columns 64 apart -> immediate-offset folds)
  for (int g = 0; g < 16; ++g) {               // uniform per wave; EXEC all-1s
    const int t0  = wave + (g << 4);
    const int cb0 = t0 << 4;
    const int col0 = cb0 + nlo;
    const float* Bbase = E + (size_t)col0 * DDIM + koff;   // tile i at +i*64*256

    v8f acc[4];
#pragma unroll
    for (int i = 0; i < 4; ++i) acc[i] = (v8f){};

#pragma unroll 8
    for (int kk = 0; kk < DDIM; kk += 4) {
      const v2f a = *(const v2f*)(Abase + kk);             // one ds_load_b64
#pragma unroll
      for (int i = 0; i < 4; ++i) {
        const v2f bv = *(const v2f*)(Bbase + (size_t)i * 64 * DDIM + kk);
        acc[i] = __builtin_amdgcn_wmma_f32_16x16x4_f32(
            /*neg_a=*/false, a, /*neg_b=*/false, bv,
            /*c_mod=*/(short)0, acc[i], /*reuse_a=*/false, /*reuse_b=*/false);
      }
    }

#pragma unroll
    for (int i = 0; i < 4; ++i) {
      const int col = col0 + i * 64;
      const float e2v = e2[col];
#pragma unroll
      for (int j = 0; j < 8; ++j) {
        const int m = j + mo;
        const float dv = z2s[m] + e2v - 2.0f * acc[i][j];
        dist_out[(size_t)(rowbase + m) * KCODES + col] = dv;
        if (dv < minv[j] || (dv == minv[j] && col < mini[j])) { minv[j] = dv; mini[j] = col; }
      }
    }
  }

  // Cross-wave argmin (first-occurrence tie-break like jnp.argmin)
#pragma unroll
  for (int j = 0; j < 8; ++j) {
    argv[wave * 256 + j * 32 + lane] = minv[j];
    argi[wave * 256 + j * 32 + lane] = mini[j];
  }
  __syncthreads();
  if (tid < 16) {
    const int j  = tid & 7;
    const int lh = (tid >> 3) * 16;              // rows 0-7 in lanes 0-15, 8-15 in 16-31
    float bv = 3.4e38f; int bi = 0x7fffffff;
    for (int w = 0; w < 4; ++w)
      for (int l = 0; l < 16; ++l) {
        const float v = argv[w * 256 + j * 32 + lh + l];
        const int   i = argi[w * 256 + j * 32 + lh + l];
        if (v < bv || (v == bv && i < bi)) { bv = v; bi = i; }
      }
    const int n = rowbase + tid;
    idx_s[tid] = bi;
    out[OUT_IDX + n] = (float)bi;
    out[OUT_ONEHOT + (size_t)n * KCODES + bi] = 1.0f;
    atomicAdd(&counts[bi], 1u);
  }
  __syncthreads();

  // zq = E[idx] written back transposed; SSE accumulated against staged z
  float sse = 0.0f;
  float* zq = out + OUT_ZQ + (size_t)b * BSTRIDE + hwbase;
  for (int i = tid; i < 4096; i += 128) {
    const int d = i >> 4, r = i & 15;
    const float ev = E[(size_t)idx_s[r] * DDIM + d];
    const float df = ev - Atile[r * 260 + d];
    sse += df * df;
    zq[(size_t)d * 1024 + r] = ev;
  }
  red[tid] = sse;
  __syncthreads();
  for (int s = 64; s > 0; s >>= 1) {             // fixed-tree -> deterministic
    if (tid < s) red[tid] += red[tid + s];
    __syncthreads();
  }
  if (tid == 0) partial[blockIdx.x] = red[0];
}

// ---------------------------------------------------------------------------
// Scalars: loss = 1.25 * mean((zq-z)^2); perplexity = exp(-sum p log(p+1e-10))
__global__ void vq_final(const float* __restrict__ partial,
                         const unsigned int* __restrict__ counts,
                         float* __restrict__ out) {
  if (threadIdx.x == 0) {
    float s = 0.0f;
    for (int i = 0; i < 2048; ++i) s += partial[i];   // in-order, deterministic
    out[OUT_LOSS] = 1.25f * (s / (float)ZELEMS);
  }
  if (threadIdx.x == 32) {
    float H = 0.0f;
    for (int k = 0; k < KCODES; ++k) {
      const float p = (float)counts[k] * (1.0f / (float)NVEC);
      H += p * logf(p + 1e-10f);
    }
    out[OUT_PERP] = expf(-H);
  }
}

// ---------------------------------------------------------------------------
extern "C" void kernel_launch(void* const* d_in, const int* in_sizes, int n_in,
                              void* d_out, int out_size, void* d_ws, size_t ws_size,
                              hipStream_t stream) {
  const float* Z = (const float*)d_in[0];   // (32,256,32,32) f32
  const float* E = (const float*)d_in[1];   // (4096,256) f32
  float* out = (float*)d_out;

  float*        partial = (float*)((char*)d_ws + WS_PARTIAL);
  unsigned int* counts  = (unsigned int*)((char*)d_ws + WS_COUNTS);
  float*        e2      = (float*)((char*)d_ws + WS_E2);

  vq_zero <<<4096, 256, 0, stream>>>(out, counts, partial);
  vq_e2   <<<KCODES / 8, 256, 0, stream>>>(E, e2);
  vq_main <<<NVEC / 16, 128, 0, stream>>>(Z, E, e2, out, counts, partial);
  vq_final<<<1, 64, 0, stream>>>(partial, counts, out);
}